// ECA_69544110457542
// MI455X (gfx1250) — compile-verified
//
#include <hip/hip_runtime.h>

// ---------------------------------------------------------------------------
// Problem constants (from the reference)
// ---------------------------------------------------------------------------
#define BB 256
#define TT 64
#define EE 256
#define XSZ 32
#define DSZ 31          // XS - 1
#define HT 80           // HS + T
#define CC_ 768         // 3*E
#define RROWS (BB * TT) // 16384
#define KK 5

typedef __attribute__((ext_vector_type(2))) float v2f;
typedef __attribute__((ext_vector_type(8))) float v8f;

// ---------------------------------------------------------------------------
// One wave32 computes a 16(M) x 64(N) fp32 output strip with
// V_WMMA_F32_16X16X4_F32.  A: 16 x K (row-major, stride lda),
// W: K x 256 (row-major), Out: 16 x (n0..n0+63) at stride ldo, bias added.
// Past-K handling: A fragment is zeroed, B row index is clamped (product = 0).
// ---------------------------------------------------------------------------
__device__ __forceinline__ void wave_gemm_16x64(
    const float* __restrict__ A, int lda, int K,
    const float* __restrict__ W,          // K x 256 row-major
    const float* __restrict__ bias,       // 256
    float* __restrict__ Out, int ldo, int n0)
{
    const int lane = threadIdx.x & 31;
    const int l16  = lane & 15;
    const int half = lane >> 4;

    v8f acc0 = {}, acc1 = {}, acc2 = {}, acc3 = {};

    const int kt_end = (K + 3) >> 2;
    for (int kt = 0; kt < kt_end; ++kt) {
        const int kbase = (kt << 2) + (half << 1);

        // A fragment: A[l16][kbase + v], zero past K
        v2f a;
        a.x = (kbase + 0 < K) ? A[l16 * lda + kbase + 0] : 0.0f;
        a.y = (kbase + 1 < K) ? A[l16 * lda + kbase + 1] : 0.0f;

        // B fragments: W[k][n0 + 16*s + l16]; clamp k past K (A is zero there)
        const int k0c = (kbase + 0 < K) ? (kbase + 0) : (K - 1);
        const int k1c = (kbase + 1 < K) ? (kbase + 1) : (K - 1);
        const float* W0 = W + k0c * 256 + n0 + l16;
        const float* W1 = W + k1c * 256 + n0 + l16;

        v2f b0, b1, b2, b3;
        b0.x = W0[0];  b0.y = W1[0];
        b1.x = W0[16]; b1.y = W1[16];
        b2.x = W0[32]; b2.y = W1[32];
        b3.x = W0[48]; b3.y = W1[48];

        acc0 = __builtin_amdgcn_wmma_f32_16x16x4_f32(false, a, false, b0,
                                                     (short)0, acc0, false, false);
        acc1 = __builtin_amdgcn_wmma_f32_16x16x4_f32(false, a, false, b1,
                                                     (short)0, acc1, false, false);
        acc2 = __builtin_amdgcn_wmma_f32_16x16x4_f32(false, a, false, b2,
                                                     (short)0, acc2, false, false);
        acc3 = __builtin_amdgcn_wmma_f32_16x16x4_f32(false, a, false, b3,
                                                     (short)0, acc3, false, false);
    }

    // D layout: lane holds D[v + 8*half][n0 + 16*s + l16] in acc_s[v]
#pragma unroll
    for (int v = 0; v < 8; ++v) {
        const int m = v + (half << 3);
        const int n = n0 + l16;
        Out[m * ldo + n +  0] = acc0[v] + bias[n +  0];
        Out[m * ldo + n + 16] = acc1[v] + bias[n + 16];
        Out[m * ldo + n + 32] = acc2[v] + bias[n + 32];
        Out[m * ldo + n + 48] = acc3[v] + bias[n + 48];
    }
}

// ---------------------------------------------------------------------------
// Kernel 0: inverse permutation (768 entries)
// ---------------------------------------------------------------------------
__global__ void ECA_invperm_kernel(const int* __restrict__ perm,
                                   int* __restrict__ inv)
{
    const int j = blockIdx.x * blockDim.x + threadIdx.x;
    if (j < CC_) inv[perm[j]] = j;
}

// ---------------------------------------------------------------------------
// Kernel 1: three input projections -> xhn[16384, 768] = [dyn | h | n]
// 12 waves/block: wave = matrix(3) x n-chunk(4), 16 rows per block.
// ---------------------------------------------------------------------------
__global__ __launch_bounds__(384) void ECA_proj_kernel(
    const float* __restrict__ x,       // [16384, 32]
    const float* __restrict__ human,   // [16384, 80]
    const float* __restrict__ nature,  // [16384, 80]
    const float* __restrict__ Wd, const float* __restrict__ bd,   // [31,256],[256]
    const float* __restrict__ Wh, const float* __restrict__ bh,   // [80,256],[256]
    const float* __restrict__ Wn, const float* __restrict__ bn,   // [80,256],[256]
    float* __restrict__ xhn)           // [16384, 768]
{
    const int r0   = blockIdx.x * 16;
    const int wave = threadIdx.x >> 5;
    const int mat  = wave >> 2;        // 0:dyn 1:h 2:n
    const int n0   = (wave & 3) * 64;

    if (mat == 0) {
        wave_gemm_16x64(x + (size_t)r0 * XSZ, XSZ, DSZ, Wd, bd,
                        xhn + (size_t)r0 * CC_ + 0, CC_, n0);
    } else if (mat == 1) {
        wave_gemm_16x64(human + (size_t)r0 * HT, HT, HT, Wh, bh,
                        xhn + (size_t)r0 * CC_ + 256, CC_, n0);
    } else {
        wave_gemm_16x64(nature + (size_t)r0 * HT, HT, HT, Wn, bn,
                        xhn + (size_t)r0 * CC_ + 512, CC_, n0);
    }
}

// ---------------------------------------------------------------------------
// Kernel 2: fused permuted depthwise conv (only the 256 needed outputs) +
// ReLU + softmax + gate, plus the value branch.  One block (256 thr) per row.
//   co[j] = conv_b[t] + sum_k xhn[perm[j-3+k]] * conv_w[t,k],  j = inv[c]
//   sel   = softmax_c(relu(co[inv[c]]))
//   cc[r, c]       = xhn[r, c] * (sel + 1) * 0.5          (c < 256)
//   cc[r, 256 + c] = x[r, 31] * Wv[c] + bv[c]
// ---------------------------------------------------------------------------
__global__ __launch_bounds__(256) void ECA_conv_softmax_kernel(
    const float* __restrict__ xhn,     // [16384, 768]
    const int*   __restrict__ perm,    // [768]
    const int*   __restrict__ inv,     // [768]
    const float* __restrict__ conv_w,  // [64, 5]
    const float* __restrict__ conv_b,  // [64]
    const float* __restrict__ x,       // [16384, 32]
    const float* __restrict__ Wv,      // [256]
    const float* __restrict__ bv,      // [256]
    float* __restrict__ cc)            // [16384, 512]
{
    __shared__ float row[CC_];
    __shared__ int   sperm[CC_];
    __shared__ float red[256];

    const int r = blockIdx.x;
    const int t = r & (TT - 1);        // r % T
    const int c = threadIdx.x;

    const float* xr = xhn + (size_t)r * CC_;
    for (int i = c; i < CC_; i += 256) {
        row[i]   = xr[i];
        sperm[i] = perm[i];
    }
    __syncthreads();

    // depthwise conv tap at output position j = inv[c] (correlation, pad 3)
    const int j = inv[c];
    const float* cw = conv_w + t * KK;
    float co = conv_b[t];
#pragma unroll
    for (int k = 0; k < KK; ++k) {
        const int p = j - 3 + k;
        if (p >= 0 && p < CC_) co += row[sperm[p]] * cw[k];
    }
    const float xx = fmaxf(co, 0.0f);

    // softmax over the 256 channels (LDS tree reduce: max, then sum)
    red[c] = xx;
    __syncthreads();
    for (int s = 128; s > 0; s >>= 1) {
        if (c < s) red[c] = fmaxf(red[c], red[c + s]);
        __syncthreads();
    }
    const float mx = red[0];
    __syncthreads();
    const float e = __expf(xx - mx);
    red[c] = e;
    __syncthreads();
    for (int s = 128; s > 0; s >>= 1) {
        if (c < s) red[c] += red[c + s];
        __syncthreads();
    }
    const float sel = e / red[0];

    const float dynv = row[c];                       // xhn[:, :256] is dynamic
    cc[(size_t)r * 512 + c]       = dynv * (sel + 1.0f) * 0.5f;
    cc[(size_t)r * 512 + 256 + c] = x[(size_t)r * XSZ + (XSZ - 1)] * Wv[c] + bv[c];
}

// ---------------------------------------------------------------------------
// Kernel 3: out[16384,256] = cc[16384,512] @ Wvd + bvd  (K = 512)
// 4 waves/block, each wave a 16x64 strip; Wvd (512KB) stays L2-resident.
// ---------------------------------------------------------------------------
__global__ __launch_bounds__(128) void ECA_out_gemm_kernel(
    const float* __restrict__ cc,      // [16384, 512]
    const float* __restrict__ Wvd,     // [512, 256]
    const float* __restrict__ bvd,     // [256]
    float* __restrict__ out)           // [16384, 256]
{
    const int r0   = blockIdx.x * 16;
    const int wave = threadIdx.x >> 5;
    wave_gemm_16x64(cc + (size_t)r0 * 512, 512, 512, Wvd, bvd,
                    out + (size_t)r0 * 256, 256, wave * 64);
}

// ---------------------------------------------------------------------------
// Host-side launcher
// ---------------------------------------------------------------------------
extern "C" void kernel_launch(void* const* d_in, const int* in_sizes, int n_in,
                              void* d_out, int out_size, void* d_ws, size_t ws_size,
                              hipStream_t stream)
{
    const float* x      = (const float*)d_in[0];
    const float* human  = (const float*)d_in[1];
    const float* nature = (const float*)d_in[2];
    const int*   perm   = (const int*)  d_in[3];
    const float* Wv     = (const float*)d_in[4];
    const float* bv     = (const float*)d_in[5];
    const float* Wd     = (const float*)d_in[6];
    const float* bd     = (const float*)d_in[7];
    const float* Wh     = (const float*)d_in[8];
    const float* bh     = (const float*)d_in[9];
    const float* Wn     = (const float*)d_in[10];
    const float* bn     = (const float*)d_in[11];
    const float* conv_w = (const float*)d_in[12];
    const float* conv_b = (const float*)d_in[13];
    const float* Wvd    = (const float*)d_in[14];
    const float* bvd    = (const float*)d_in[15];
    float* out = (float*)d_out;

    // workspace layout: [inv: 768 ints (pad 4KB)] [xhn: 16384*768 f32] [cc: 16384*512 f32]
    int*   inv = (int*)d_ws;
    float* xhn = (float*)((char*)d_ws + 4096);
    float* cc  = xhn + (size_t)RROWS * CC_;

    ECA_invperm_kernel<<<3, 256, 0, stream>>>(perm, inv);

    ECA_proj_kernel<<<RROWS / 16, 384, 0, stream>>>(
        x, human, nature, Wd, bd, Wh, bh, Wn, bn, xhn);

    ECA_conv_softmax_kernel<<<RROWS, 256, 0, stream>>>(
        xhn, perm, inv, conv_w, conv_b, x, Wv, bv, cc);

    ECA_out_gemm_kernel<<<RROWS / 16, 128, 0, stream>>>(cc, Wvd, bvd, out);

    (void)in_sizes; (void)n_in; (void)out_size; (void)ws_size;
}